// SelfAttentionLayer_28372553957392
// MI455X (gfx1250) — compile-verified
//
#include <hip/hip_runtime.h>
#include <hip/hip_bf16.h>
#include <math.h>

// ---------------------------------------------------------------------------
// Transformer encoder layer for MI455X (gfx1250, wave32, WMMA).
// fp32 V_WMMA_F32_16X16X4_F32 throughout: the layer is bandwidth-bound
// (attn tensor alone is 134MB), so full-precision matrix ops are free.
// 2x2 register blocking doubles FLOP/byte on the vector-load path; the AV
// kernel blocks the whole head dim so attn is read exactly once.
// ---------------------------------------------------------------------------

typedef float v2f __attribute__((ext_vector_type(2)));
typedef float v8f __attribute__((ext_vector_type(8)));

#define WMMA_F32x4(acc, a, b)                                                  \
  acc = __builtin_amdgcn_wmma_f32_16x16x4_f32(false, (a), false, (b),          \
                                              (short)0, (acc), false, false)

static constexpr int Bb = 2;
static constexpr int Ss = 1024;
static constexpr int Dd = 1024;
static constexpr int Hh = 16;
static constexpr int DhC = 64;
static constexpr int Mrows = Bb * Ss;            // 2048 token rows
static constexpr size_t MD = (size_t)Mrows * Dd; // elems in a [B*S, D] buffer

// ---------------------------------------------------------------------------
// NT GEMM, 2x2 register-blocked:  C[M,N] = A[M,K] * Bw[N,K]^T (+bias)(relu)
// One wave computes a 32x32 C tile (4 f32 WMMA accumulators).
//
// f32 WMMA 16x16x4 operand layout (wave32):
//   A (16x4):  lane l -> row m = l&15 ; VGPR0/1 = K = 2*(l>>4) + {0,1}
//   B (4x16):  lane l -> col n = l&15 ; VGPR0/1 = K = 2*(l>>4) + {0,1}
//   C (16x16): VGPR r -> row m = r + 8*(l>>4), col n = l&15
// Each fragment is one aligned 8-byte load per lane; each fragment feeds
// two WMMAs (8 FLOP/byte from the load pipe).
// ---------------------------------------------------------------------------
template <bool BIAS, bool RELU>
__global__ __launch_bounds__(256) void gemm_nt_wmma(
    const float* __restrict__ A, const float* __restrict__ Bw,
    const float* __restrict__ bias, float* __restrict__ C,
    int M, int N, int K) {
  const int lane = threadIdx.x & 31;
  const int wave = threadIdx.x >> 5;
  const int tilesN = N >> 5;
  const int tile = blockIdx.x * (blockDim.x >> 5) + wave;
  const int tm = tile / tilesN;
  const int tn = tile % tilesN;
  if (tm * 32 >= M) return;  // wave-uniform guard: EXEC stays all-ones

  const int r  = lane & 15;        // row within a 16-wide fragment
  const int kh = (lane >> 4) << 1; // 0 or 2 : K sub-offset for this half-wave

  const float* __restrict__ a0 = A  + (size_t)(tm * 32 + r)      * K + kh;
  const float* __restrict__ a1 = A  + (size_t)(tm * 32 + 16 + r) * K + kh;
  const float* __restrict__ b0 = Bw + (size_t)(tn * 32 + r)      * K + kh;
  const float* __restrict__ b1 = Bw + (size_t)(tn * 32 + 16 + r) * K + kh;

  v8f acc00 = {}, acc01 = {}, acc10 = {}, acc11 = {};
  for (int k0 = 0; k0 < K; k0 += 16) {
    if (k0 + 64 < K) {  // uniform: stream next cachelines toward L0
      __builtin_prefetch(a0 + k0 + 64, 0, 0);
      __builtin_prefetch(a1 + k0 + 64, 0, 0);
      __builtin_prefetch(b0 + k0 + 64, 0, 0);
      __builtin_prefetch(b1 + k0 + 64, 0, 0);
    }
#pragma unroll
    for (int u = 0; u < 4; ++u) {
      v2f fa0 = *(const v2f*)(a0 + k0 + u * 4);
      v2f fa1 = *(const v2f*)(a1 + k0 + u * 4);
      v2f fb0 = *(const v2f*)(b0 + k0 + u * 4);
      v2f fb1 = *(const v2f*)(b1 + k0 + u * 4);
      WMMA_F32x4(acc00, fa0, fb0);
      WMMA_F32x4(acc01, fa0, fb1);
      WMMA_F32x4(acc10, fa1, fb0);
      WMMA_F32x4(acc11, fa1, fb1);
    }
  }

  const int n0 = tn * 32 + (lane & 15);
  const int n1 = n0 + 16;
  const int m0 = tm * 32 + ((lane >> 4) << 3);
  const int m1 = m0 + 16;
  const float bv0 = BIAS ? bias[n0] : 0.0f;
  const float bv1 = BIAS ? bias[n1] : 0.0f;
#pragma unroll
  for (int rr = 0; rr < 8; ++rr) {
    float v00 = acc00[rr] + bv0, v01 = acc01[rr] + bv1;
    float v10 = acc10[rr] + bv0, v11 = acc11[rr] + bv1;
    if (RELU) {
      v00 = fmaxf(v00, 0.0f); v01 = fmaxf(v01, 0.0f);
      v10 = fmaxf(v10, 0.0f); v11 = fmaxf(v11, 0.0f);
    }
    C[(size_t)(m0 + rr) * N + n0] = v00;
    C[(size_t)(m0 + rr) * N + n1] = v01;
    C[(size_t)(m1 + rr) * N + n0] = v10;
    C[(size_t)(m1 + rr) * N + n1] = v11;
  }
}

// ---------------------------------------------------------------------------
// Attention scores, 2x2 blocked: 32x32 (q,k) tile per wave, K-dim = Dh = 64.
// attn[b,h,q,k] = (pad[b,k] ? -1e12 : Q.Kh) * (1/sqrt(Dh))
// Mask applied BEFORE the scale, matching the reference.
// ---------------------------------------------------------------------------
__global__ __launch_bounds__(256) void attn_scores_wmma(
    const float* __restrict__ Q, const float* __restrict__ Kmat,
    const int* __restrict__ mask, float* __restrict__ attn) {
  const int lane = threadIdx.x & 31;
  const int wave = threadIdx.x >> 5;
  const int tile = blockIdx.x * (blockDim.x >> 5) + wave;

  const int tiles32 = Ss >> 5;               // 32
  const int tk = tile % tiles32;
  const int tq = (tile / tiles32) % tiles32;
  const int bh = tile / (tiles32 * tiles32); // 0..31
  const int b  = bh / Hh;
  const int h  = bh % Hh;

  const int r  = lane & 15;
  const int kh = (lane >> 4) << 1;

  const float* __restrict__ q0 =
      Q + ((size_t)(b * Ss + tq * 32 + r)      * Dd) + h * DhC + kh;
  const float* __restrict__ q1 =
      Q + ((size_t)(b * Ss + tq * 32 + 16 + r) * Dd) + h * DhC + kh;
  const float* __restrict__ k0p =
      Kmat + ((size_t)(b * Ss + tk * 32 + r)      * Dd) + h * DhC + kh;
  const float* __restrict__ k1p =
      Kmat + ((size_t)(b * Ss + tk * 32 + 16 + r) * Dd) + h * DhC + kh;

  v8f acc00 = {}, acc01 = {}, acc10 = {}, acc11 = {};
#pragma unroll
  for (int k0 = 0; k0 < DhC; k0 += 16) {
#pragma unroll
    for (int u = 0; u < 4; ++u) {
      v2f fa0 = *(const v2f*)(q0  + k0 + u * 4);
      v2f fa1 = *(const v2f*)(q1  + k0 + u * 4);
      v2f fb0 = *(const v2f*)(k0p + k0 + u * 4);
      v2f fb1 = *(const v2f*)(k1p + k0 + u * 4);
      WMMA_F32x4(acc00, fa0, fb0);
      WMMA_F32x4(acc01, fa0, fb1);
      WMMA_F32x4(acc10, fa1, fb0);
      WMMA_F32x4(acc11, fa1, fb1);
    }
  }

  const int kc0 = tk * 32 + (lane & 15);
  const int kc1 = kc0 + 16;
  const int qb0 = tq * 32 + ((lane >> 4) << 3);
  const int qb1 = qb0 + 16;
  const bool pad0 = mask[b * Ss + kc0] != 0;
  const bool pad1 = mask[b * Ss + kc1] != 0;
  float* __restrict__ abase = attn + ((size_t)bh * Ss) * Ss;
#pragma unroll
  for (int rr = 0; rr < 8; ++rr) {
    abase[(size_t)(qb0 + rr) * Ss + kc0] = (pad0 ? -1.0e12f : acc00[rr]) * 0.125f;
    abase[(size_t)(qb0 + rr) * Ss + kc1] = (pad1 ? -1.0e12f : acc01[rr]) * 0.125f;
    abase[(size_t)(qb1 + rr) * Ss + kc0] = (pad0 ? -1.0e12f : acc10[rr]) * 0.125f;
    abase[(size_t)(qb1 + rr) * Ss + kc1] = (pad1 ? -1.0e12f : acc11[rr]) * 0.125f;
  }
}

// ---------------------------------------------------------------------------
// Row softmax over the last axis (1024), in place. One 256-thread block/row.
// ---------------------------------------------------------------------------
__global__ __launch_bounds__(256) void softmax_rows(float* __restrict__ attn) {
  const int t    = threadIdx.x;
  const int lane = t & 31;
  const int w    = t >> 5;
  float* __restrict__ p = attn + (size_t)blockIdx.x * Ss;

  float v[4];
#pragma unroll
  for (int i = 0; i < 4; ++i) v[i] = p[t + i * 256];

  __shared__ float red[8];

  // --- max ---
  float m = fmaxf(fmaxf(v[0], v[1]), fmaxf(v[2], v[3]));
#pragma unroll
  for (int off = 16; off > 0; off >>= 1) m = fmaxf(m, __shfl_xor(m, off, 32));
  if (lane == 0) red[w] = m;
  __syncthreads();
  m = red[0];
#pragma unroll
  for (int i = 1; i < 8; ++i) m = fmaxf(m, red[i]);
  __syncthreads();

  // --- exp + sum ---
  float s = 0.0f;
#pragma unroll
  for (int i = 0; i < 4; ++i) {
    v[i] = expf(v[i] - m);
    s += v[i];
  }
#pragma unroll
  for (int off = 16; off > 0; off >>= 1) s += __shfl_xor(s, off, 32);
  if (lane == 0) red[w] = s;
  __syncthreads();
  float tot = red[0];
#pragma unroll
  for (int i = 1; i < 8; ++i) tot += red[i];

  const float inv = 1.0f / tot;
#pragma unroll
  for (int i = 0; i < 4; ++i) p[t + i * 256] = v[i] * inv;
}

// ---------------------------------------------------------------------------
// AV: av[b,q,h,d] = sum_k attn[b,h,q,k] * V[b,k,h,d].
// One wave computes the FULL head dim (16 q-rows x 64 d-cols, 4 accumulators)
// so every attn element (the 134MB tensor) is read exactly once.
// attn is K-contiguous (v2f loads); V columns strided by D.
// ---------------------------------------------------------------------------
__global__ __launch_bounds__(256) void attn_av_wmma(
    const float* __restrict__ attn, const float* __restrict__ V,
    float* __restrict__ AV) {
  const int lane = threadIdx.x & 31;
  const int wave = threadIdx.x >> 5;
  const int tile = blockIdx.x * (blockDim.x >> 5) + wave;

  const int tq = tile % (Ss >> 4);
  const int bh = tile / (Ss >> 4);
  const int b  = bh / Hh;
  const int h  = bh % Hh;

  const int r  = lane & 15;
  const int kh = (lane >> 4) << 1;

  const float* __restrict__ a_ptr =
      attn + ((size_t)bh * Ss + tq * 16 + r) * Ss + kh;
  // per-lane column pointers into V, one per 16-wide d fragment
  const float* __restrict__ vc = V + (size_t)(b * Ss) * Dd + h * DhC + (lane & 15);

  v8f acc0 = {}, acc1 = {}, acc2 = {}, acc3 = {};
  for (int k0 = 0; k0 < Ss; k0 += 16) {
#pragma unroll
    for (int u = 0; u < 4; ++u) {
      const int k = k0 + u * 4 + kh;
      v2f a = *(const v2f*)(a_ptr + k0 + u * 4);
      const float* __restrict__ vr0 = vc + (size_t)k * Dd;
      const float* __restrict__ vr1 = vc + (size_t)(k + 1) * Dd;
      v2f f0, f1, f2, f3;
      f0.x = vr0[0];  f0.y = vr1[0];
      f1.x = vr0[16]; f1.y = vr1[16];
      f2.x = vr0[32]; f2.y = vr1[32];
      f3.x = vr0[48]; f3.y = vr1[48];
      WMMA_F32x4(acc0, a, f0);
      WMMA_F32x4(acc1, a, f1);
      WMMA_F32x4(acc2, a, f2);
      WMMA_F32x4(acc3, a, f3);
    }
  }

  const int n     = lane & 15;
  const int qbase = tq * 16 + ((lane >> 4) << 3);
#pragma unroll
  for (int rr = 0; rr < 8; ++rr) {
    float* __restrict__ orow = AV + (size_t)(b * Ss + qbase + rr) * Dd + h * DhC + n;
    orow[0]  = acc0[rr];
    orow[16] = acc1[rr];
    orow[32] = acc2[rr];
    orow[48] = acc3[rr];
  }
}

// ---------------------------------------------------------------------------
// out = LayerNorm(a + b) with unbiased std (ddof=1) and eps added to std,
// exactly matching the reference. One 256-thread block per token row.
// ---------------------------------------------------------------------------
__global__ __launch_bounds__(256) void residual_ln(
    const float* __restrict__ a, const float* __restrict__ b,
    const float* __restrict__ gamma, const float* __restrict__ beta,
    float* __restrict__ out) {
  const int t    = threadIdx.x;
  const int lane = t & 31;
  const int w    = t >> 5;
  const size_t row = blockIdx.x;

  float v[4];
#pragma unroll
  for (int i = 0; i < 4; ++i) {
    const int c = t + i * 256;
    v[i] = a[row * Dd + c] + b[row * Dd + c];
  }

  __shared__ float red[8];

  float s = v[0] + v[1] + v[2] + v[3];
#pragma unroll
  for (int off = 16; off > 0; off >>= 1) s += __shfl_xor(s, off, 32);
  if (lane == 0) red[w] = s;
  __syncthreads();
  float tot = red[0];
#pragma unroll
  for (int i = 1; i < 8; ++i) tot += red[i];
  const float mean = tot * (1.0f / 1024.0f);
  __syncthreads();

  float q = 0.0f;
#pragma unroll
  for (int i = 0; i < 4; ++i) {
    const float d = v[i] - mean;
    q += d * d;
  }
#pragma unroll
  for (int off = 16; off > 0; off >>= 1) q += __shfl_xor(q, off, 32);
  if (lane == 0) red[w] = q;
  __syncthreads();
  float qtot = red[0];
#pragma unroll
  for (int i = 1; i < 8; ++i) qtot += red[i];

  const float stdv = sqrtf(qtot * (1.0f / 1023.0f));  // ddof=1
  const float inv  = 1.0f / (stdv + 1e-8f);           // eps on std
#pragma unroll
  for (int i = 0; i < 4; ++i) {
    const int c = t + i * 256;
    out[row * Dd + c] = gamma[c] * (v[i] - mean) * inv + beta[c];
  }
}

// ---------------------------------------------------------------------------
extern "C" void kernel_launch(void* const* d_in, const int* in_sizes, int n_in,
                              void* d_out, int out_size, void* d_ws,
                              size_t ws_size, hipStream_t stream) {
  (void)in_sizes; (void)n_in; (void)out_size; (void)ws_size;

  const float* x     = (const float*)d_in[0];
  const int*   mask  = (const int*)d_in[1];
  const float* Wq    = (const float*)d_in[2];
  const float* Wk    = (const float*)d_in[3];
  const float* Wv    = (const float*)d_in[4];
  const float* Wo    = (const float*)d_in[5];
  const float* W1    = (const float*)d_in[6];
  const float* b1    = (const float*)d_in[7];
  const float* W2    = (const float*)d_in[8];
  const float* b2    = (const float*)d_in[9];
  const float* g1    = (const float*)d_in[10];
  const float* be1   = (const float*)d_in[11];
  const float* g2    = (const float*)d_in[12];
  const float* be2   = (const float*)d_in[13];

  float* out  = (float*)d_out;           // [B,S,D]
  float* attn = out + MD;                // [B,H,S,S] (tuple output #2)

  float* buf0 = (float*)d_ws;            // Q  -> later attn_out -> later h2
  float* buf1 = buf0 + MD;               // K  -> later x1
  float* buf2 = buf1 + MD;               // V  -> later h1
  float* buf3 = buf2 + MD;               // av

  const dim3 blk(256);
  // 32x32 wave-tiles, 8 waves per block
  const int gemmBlocks  = (Mrows / 32) * (Dd / 32) / 8;             // 256
  const int scoreBlocks = Bb * Hh * (Ss / 32) * (Ss / 32) / 8;      // 4096
  const int softBlocks  = Bb * Hh * Ss;                             // 32768
  const int avBlocks    = Bb * Hh * (Ss / 16) / 8;                  // 256
  const int lnBlocks    = Mrows;                                    // 2048

  // 1) Q, K, V projections
  gemm_nt_wmma<false, false><<<gemmBlocks, blk, 0, stream>>>(x, Wq, nullptr, buf0, Mrows, Dd, Dd);
  gemm_nt_wmma<false, false><<<gemmBlocks, blk, 0, stream>>>(x, Wk, nullptr, buf1, Mrows, Dd, Dd);
  gemm_nt_wmma<false, false><<<gemmBlocks, blk, 0, stream>>>(x, Wv, nullptr, buf2, Mrows, Dd, Dd);

  // 2) masked, scaled scores -> attn region of d_out
  attn_scores_wmma<<<scoreBlocks, blk, 0, stream>>>(buf0, buf1, mask, attn);

  // 3) softmax in place (this IS the second tuple output)
  softmax_rows<<<softBlocks, blk, 0, stream>>>(attn);

  // 4) av = attn @ V   (buf3)
  attn_av_wmma<<<avBlocks, blk, 0, stream>>>(attn, buf2, buf3);

  // 5) attn_out = av @ Wo^T  (reuse buf0; Q dead)
  gemm_nt_wmma<false, false><<<gemmBlocks, blk, 0, stream>>>(buf3, Wo, nullptr, buf0, Mrows, Dd, Dd);

  // 6) x1 = LN(x + attn_out)  (reuse buf1; K dead)
  residual_ln<<<lnBlocks, blk, 0, stream>>>(x, buf0, g1, be1, buf1);

  // 7) h1 = relu(x1 @ W1^T + b1)  (reuse buf2; V dead)
  gemm_nt_wmma<true, true><<<gemmBlocks, blk, 0, stream>>>(buf1, W1, b1, buf2, Mrows, Dd, Dd);

  // 8) h2 = h1 @ W2^T + b2  (reuse buf0; attn_out dead)
  gemm_nt_wmma<true, false><<<gemmBlocks, blk, 0, stream>>>(buf2, W2, b2, buf0, Mrows, Dd, Dd);

  // 9) out = LN(x1 + h2)
  residual_ln<<<lnBlocks, blk, 0, stream>>>(buf1, buf0, g2, be2, out);
}